// Quantizer_51711406244033
// MI455X (gfx1250) — compile-verified
//
#include <hip/hip_runtime.h>

#define DIM     512
#define KCODES  256
#define CBOOKS  16
#define BATCH   16384
#define SUBDIM  32      // DIM / CBOOKS
#define TILE_B  128     // batch rows per block (8 waves x 16 rows)
#define WSTRIDE 34      // padded LDS stride (even -> 8B aligned, conflict-free)
#define XSTRIDE 34

typedef float v2f __attribute__((ext_vector_type(2)));
typedef float v8f __attribute__((ext_vector_type(8)));

__global__ __launch_bounds__(256) void quantizer_main(
    const float* __restrict__ x, const float* __restrict__ weight,
    const float* __restrict__ bias, const float* __restrict__ to_output,
    float* __restrict__ partials)
{
    __shared__ float lds_w[KCODES * WSTRIDE];   // codebook panel  (256 x 32, padded)
    __shared__ float lds_x[TILE_B * XSTRIDE];   // x slice         (128 x 32, padded)
    __shared__ float lds_bias[KCODES];
    __shared__ int   lds_idx[TILE_B];
    __shared__ float lds_red[16];

    const int tid   = threadIdx.x;
    const int lane  = tid & 31;
    const int wid   = tid >> 5;       // wave id 0..7
    const int khalf = lane >> 4;      // lane half selects K pair (A/B frags)
    const int l16   = lane & 15;
    const int b0    = blockIdx.x * TILE_B;

    float err_acc = 0.0f;
    float x2_acc  = 0.0f;

    for (int c = 0; c < CBOOKS; ++c) {
        // ---- stage codebook panel, bias, and x slice into LDS (coalesced) ----
        {
            const int col   = tid & 31;
            const int rbase = tid >> 5;   // 0..7, 8 rows per sweep
            for (int rr = 0; rr < KCODES; rr += 8) {
                const int row = rr + rbase;
                lds_w[row * WSTRIDE + col] =
                    weight[(size_t)(c * KCODES + row) * DIM + c * SUBDIM + col];
            }
            for (int rr = 0; rr < TILE_B; rr += 8) {
                const int row = rr + rbase;
                lds_x[row * XSTRIDE + col] =
                    x[(size_t)(b0 + row) * DIM + c * SUBDIM + col];
            }
            if (tid < KCODES) lds_bias[tid] = bias[c * KCODES + tid];
        }
        __syncthreads();

        // ---- A fragments: this wave's 16 rows x 32 K, 8 chunks of K=4 ----
        // 32-bit A 16x4 layout: lane = (k>=2)*16 + m, vgpr = k&1
        v2f a[8];
        #pragma unroll
        for (int j = 0; j < 8; ++j) {
            const int k = 4 * j + 2 * khalf;
            const float* p = &lds_x[(wid * 16 + l16) * XSTRIDE + k];
            a[j].x = p[0];
            a[j].y = p[1];
        }

        float best_val[8];
        int   best_idx[8];
        #pragma unroll
        for (int r = 0; r < 8; ++r) { best_val[r] = -3.4e38f; best_idx[r] = 0; }

        // ---- 16 N-tiles of 16 codes; each = 8 chained f32 WMMAs over K=32 ----
        for (int t = 0; t < 16; ++t) {
            v8f acc = {};
            #pragma unroll
            for (int j = 0; j < 8; ++j) {
                const int k = 4 * j + 2 * khalf;
                // B 4x16 layout (mirror of A): lane = n + (k>=2)*16, vgpr = k&1
                const float* p = &lds_w[(t * 16 + l16) * WSTRIDE + k];
                v2f b;
                b.x = p[0];
                b.y = p[1];
                acc = __builtin_amdgcn_wmma_f32_16x16x4_f32(
                          false, a[j], false, b, (short)0, acc, false, false);
            }
            // C/D layout: lanes 0-15 N=lane (M=r), lanes 16-31 N=lane-16 (M=r+8)
            const float bv = lds_bias[t * 16 + l16];
            const int   n  = t * 16 + l16;
            #pragma unroll
            for (int r = 0; r < 8; ++r) {
                const float v = acc[r] + bv;   // LOGITS_SCALE>0: argmax-invariant
                if (v > best_val[r]) { best_val[r] = v; best_idx[r] = n; }
            }
        }

        // ---- argmax across the 16 lanes of each half (ties -> lowest idx) ----
        #pragma unroll
        for (int s = 1; s < 16; s <<= 1) {
            #pragma unroll
            for (int r = 0; r < 8; ++r) {
                const float ov = __shfl_xor(best_val[r], s, 16);
                const int   oi = __shfl_xor(best_idx[r], s, 16);
                if (ov > best_val[r] || (ov == best_val[r] && oi < best_idx[r])) {
                    best_val[r] = ov; best_idx[r] = oi;
                }
            }
        }
        if (l16 == 0) {
            #pragma unroll
            for (int r = 0; r < 8; ++r)
                lds_idx[wid * 16 + khalf * 8 + r] = best_idx[r];
        }
        __syncthreads();

        // ---- error accumulation: (to_output[chosen] - x)^2 and x^2 ----
        #pragma unroll
        for (int m = 0; m < 16; ++m) {
            const int   idx = lds_idx[wid * 16 + m];
            const float wv  = to_output[(size_t)(c * KCODES + idx) * DIM
                                        + c * SUBDIM + lane];
            const float xv  = lds_x[(wid * 16 + m) * XSTRIDE + lane];
            const float e   = wv - xv;
            err_acc += e * e;
            x2_acc  += xv * xv;
        }
        __syncthreads();   // protect LDS before next codebook's staging
    }

    // ---- block reduction ----
    #pragma unroll
    for (int s = 1; s < 32; s <<= 1) {
        err_acc += __shfl_xor(err_acc, s, 32);
        x2_acc  += __shfl_xor(x2_acc,  s, 32);
    }
    if (lane == 0) { lds_red[wid] = err_acc; lds_red[8 + wid] = x2_acc; }
    __syncthreads();
    if (tid == 0) {
        float e = 0.0f, xs = 0.0f;
        #pragma unroll
        for (int w = 0; w < 8; ++w) { e += lds_red[w]; xs += lds_red[8 + w]; }
        partials[blockIdx.x * 2 + 0] = e;
        partials[blockIdx.x * 2 + 1] = xs;
    }
}

__global__ void quantizer_finalize(const float* __restrict__ partials,
                                   float* __restrict__ out, int nblk)
{
    if (blockIdx.x == 0 && threadIdx.x == 0) {
        double e = 0.0, xs = 0.0;
        for (int i = 0; i < nblk; ++i) {
            e  += (double)partials[2 * i + 0];
            xs += (double)partials[2 * i + 1];
        }
        out[0] = (float)(e / (xs + 1e-20));
    }
}

extern "C" void kernel_launch(void* const* d_in, const int* in_sizes, int n_in,
                              void* d_out, int out_size, void* d_ws, size_t ws_size,
                              hipStream_t stream)
{
    const float* x         = (const float*)d_in[0];
    const float* weight    = (const float*)d_in[1];
    const float* bias      = (const float*)d_in[2];
    const float* to_output = (const float*)d_in[3];
    // d_in[4] = mask: recomputed implicitly via index arithmetic.

    float* partials = (float*)d_ws;   // 2 floats per block
    float* out      = (float*)d_out;

    const int nblk = BATCH / TILE_B;  // 128
    quantizer_main<<<nblk, 256, 0, stream>>>(x, weight, bias, to_output, partials);
    quantizer_finalize<<<1, 32, 0, stream>>>(partials, out, nblk);
}